// ApproximatePHProbe_76441827934265
// MI455X (gfx1250) — compile-verified
//
#include <hip/hip_runtime.h>
#include <hip/hip_bf16.h>
#include <math.h>

// ---------------------------------------------------------------------------
// ApproximatePHProbe on MI455X (gfx1250)
//   Stage 1: farthest-point sampling (255 sequential L2-resident sweeps)
//   Stage 2: 256x256 landmark distance matrix via V_WMMA_F32_16X16X4_F32
//   Stage 3: one-WGP parallel Jacobi eigensolver, 256KB matrix in LDS
// ---------------------------------------------------------------------------

#define DIM 64
#define NLM 256
#define NSIG 10
#define BLK 256
#define NEG_INF (-3.402823466e+38f)

typedef float v2f __attribute__((ext_vector_type(2)));
typedef float v8f __attribute__((ext_vector_type(8)));

// ---------------------------------------------------------------------------
// Init: dists = +inf, landmark 0 = points[0]
// ---------------------------------------------------------------------------
__global__ void fps_init_kernel(const float* __restrict__ points, int n,
                                float* __restrict__ dists,
                                float* __restrict__ lm0) {
  int i = blockIdx.x * BLK + threadIdx.x;
  if (i < n) dists[i] = 3.402823466e+38f;
  if (blockIdx.x == 0 && threadIdx.x < DIM / 4)
    ((float4*)lm0)[threadIdx.x] = ((const float4*)points)[threadIdx.x];
}

// ---------------------------------------------------------------------------
// FPS sweep: dists[i] = min(dists[i], dist(points[i], lmk)), partial argmax.
// 500k x 64 f32 per sweep -> L2-resident streaming, bandwidth bound.
// ---------------------------------------------------------------------------
__global__ void fps_update_kernel(const float* __restrict__ points, int n,
                                  const float* __restrict__ lmk,
                                  float* __restrict__ dists,
                                  float* __restrict__ pmax,
                                  int* __restrict__ pidx) {
  __shared__ float4 slm[DIM / 4];
  __shared__ float sval[BLK];
  __shared__ int   sid[BLK];
  int tid = threadIdx.x;
  if (tid < DIM / 4) slm[tid] = ((const float4*)lmk)[tid];
  __syncthreads();

  int i = blockIdx.x * BLK + tid;
  float best = NEG_INF;
  if (i < n) {
    const float4* p = (const float4*)(points + (size_t)i * DIM);
    float sq = 0.0f;
#pragma unroll
    for (int j = 0; j < DIM / 4; ++j) {
      float4 v = p[j];
      float4 l = slm[j];
      float dx = v.x - l.x, dy = v.y - l.y, dz = v.z - l.z, dw = v.w - l.w;
      sq += dx * dx + dy * dy + dz * dz + dw * dw;
    }
    float d  = sq > 0.0f ? sqrtf(sq) : 0.0f;   // grad-safe dist semantics
    float nd = fminf(dists[i], d);
    dists[i] = nd;
    best = nd;
  }
  sval[tid] = best;
  sid[tid]  = i;
  __syncthreads();
  for (int s = BLK / 2; s > 0; s >>= 1) {
    if (tid < s) {
      float a = sval[tid], b = sval[tid + s];
      if (b > a || (b == a && sid[tid + s] < sid[tid])) {  // argmax, first-index tie
        sval[tid] = b;
        sid[tid]  = sid[tid + s];
      }
    }
    __syncthreads();
  }
  if (tid == 0) { pmax[blockIdx.x] = sval[0]; pidx[blockIdx.x] = sid[0]; }
}

// ---------------------------------------------------------------------------
// Finalize argmax over block partials; gather next landmark's 64 coords.
// ---------------------------------------------------------------------------
__global__ void fps_select_kernel(const float* __restrict__ points,
                                  const float* __restrict__ pmax,
                                  const int* __restrict__ pidx, int nb,
                                  float* __restrict__ lm_out) {
  __shared__ float sval[BLK];
  __shared__ int   sid[BLK];
  int tid = threadIdx.x;
  float best = NEG_INF;
  int bidx = 0x7fffffff;
  for (int b = tid; b < nb; b += BLK) {
    float v = pmax[b];
    int   ix = pidx[b];
    if (v > best || (v == best && ix < bidx)) { best = v; bidx = ix; }
  }
  sval[tid] = best;
  sid[tid]  = bidx;
  __syncthreads();
  for (int s = BLK / 2; s > 0; s >>= 1) {
    if (tid < s) {
      float a = sval[tid], b = sval[tid + s];
      if (b > a || (b == a && sid[tid + s] < sid[tid])) {
        sval[tid] = b;
        sid[tid]  = sid[tid + s];
      }
    }
    __syncthreads();
  }
  int far = sid[0];
  if (tid < DIM / 4)
    ((float4*)lm_out)[tid] = ((const float4*)(points + (size_t)far * DIM))[tid];
}

// ---------------------------------------------------------------------------
// Landmark distance matrix via f32 WMMA:
//   G = LM * LM^T  (256x256x64) with V_WMMA_F32_16X16X4_F32
//   D[i][j] = sqrt(max(|pi|^2+|pj|^2-2G, 0)), diag = 1e-6
// One workgroup (8 waves), 256 output tiles of 16x16, operands in LDS.
// Dynamic LDS: 16384 floats (landmarks) + 256 floats (norms).
// ---------------------------------------------------------------------------
__global__ void gram_wmma_kernel(const float* __restrict__ lm,
                                 float* __restrict__ D) {
  extern __shared__ float smem[];
  float* sA = smem;            // [256][64]
  float* sn = smem + NLM * DIM;  // [256]
  int tid = threadIdx.x;

  for (int i = tid; i < NLM * DIM / 4; i += BLK)
    ((float4*)sA)[i] = ((const float4*)lm)[i];
  __syncthreads();

  float acc = 0.0f;
#pragma unroll 8
  for (int j = 0; j < DIM; ++j) {
    float v = sA[tid * DIM + j];
    acc += v * v;
  }
  sn[tid] = acc;
  __syncthreads();

  int wave = tid >> 5;       // 0..7
  int lane = tid & 31;
  int hl   = lane >> 4;      // 0 or 1 (ISA lane-half split)
  int r    = lane & 15;

  for (int t = wave; t < 256; t += 8) {       // 16x16 grid of 16x16 tiles
    int ti = t >> 4, tj = t & 15;
    v8f c = {};
#pragma unroll
    for (int kk = 0; kk < DIM / 4; ++kk) {
      int k0 = kk * 4 + 2 * hl;               // A 16x4 f32 layout: K = 2*half + vgpr
      v2f a, b;
      a.x = sA[(ti * 16 + r) * DIM + k0];
      a.y = sA[(ti * 16 + r) * DIM + k0 + 1];
      b.x = sA[(tj * 16 + r) * DIM + k0];     // B = A^T tile (Gram)
      b.y = sA[(tj * 16 + r) * DIM + k0 + 1];
      c = __builtin_amdgcn_wmma_f32_16x16x4_f32(
          /*neg_a=*/false, a, /*neg_b=*/false, b,
          /*c_mod=*/(short)0, c, /*reuse_a=*/false, /*reuse_b=*/false);
    }
#pragma unroll
    for (int v = 0; v < 8; ++v) {             // C layout: VGPR v -> M = v + 8*half
      int M = ti * 16 + v + 8 * hl;
      int N = tj * 16 + r;
      float sq = sn[M] + sn[N] - 2.0f * c[v];
      sq = sq > 0.0f ? sq : 0.0f;
      D[M * NLM + N] = (M == N) ? 1e-6f : sqrtf(sq);
    }
  }
}

// ---------------------------------------------------------------------------
// Parallel cyclic Jacobi eigensolver, one workgroup, matrix in 256KB LDS
// (CDNA5's 320KB/WGP makes this possible). 128 disjoint pairs/round via
// closed-form round-robin; 255 rounds/sweep, 8 sweeps. Then top-10 diag,
// normalize, rupture test -> out[0..11].
// ---------------------------------------------------------------------------
__global__ void jacobi_kernel(const float* __restrict__ Dm,
                              const float* __restrict__ prev_sig,
                              const int* __restrict__ step_count,
                              float* __restrict__ out) {
  extern __shared__ float A[];              // 65536 floats = 256 KB
  __shared__ float sc[128], ss[128];
  __shared__ int   sp[128], sqi[128];
  __shared__ float ev[NLM];
  int tid = threadIdx.x;

  for (int i = tid; i < NLM * NLM / 4; i += BLK)
    ((float4*)A)[i] = ((const float4*)Dm)[i];
  __syncthreads();

  const int NSWEEP = 8;
  for (int sweep = 0; sweep < NSWEEP; ++sweep) {
    for (int rr = 0; rr < NLM - 1; ++rr) {
      if (tid < 128) {
        int p, q;
        if (tid == 0) { p = 255; q = rr % 255; }
        else { p = (rr + tid) % 255; q = (rr + 255 - tid) % 255; }
        if (p > q) { int t2 = p; p = q; q = t2; }
        float app = A[p * NLM + p], aqq = A[q * NLM + q], apq = A[p * NLM + q];
        float c = 1.0f, s = 0.0f;
        if (fabsf(apq) > 1e-12f) {
          float tau = (aqq - app) / (2.0f * apq);
          float t = (tau >= 0.0f ? 1.0f : -1.0f) /
                    (fabsf(tau) + sqrtf(1.0f + tau * tau));
          c = 1.0f / sqrtf(1.0f + t * t);
          s = t * c;
        }
        sc[tid] = c; ss[tid] = s; sp[tid] = p; sqi[tid] = q;
      }
      __syncthreads();
      // A <- A * J   (disjoint column pairs, no conflicts)
      for (int it = 0; it < 128; ++it) {
        int idx = it * BLK + tid;
        int k = idx >> 8, i = idx & 255;
        int p = sp[k], q = sqi[k];
        float c = sc[k], s = ss[k];
        float aip = A[i * NLM + p], aiq = A[i * NLM + q];
        A[i * NLM + p] = c * aip - s * aiq;
        A[i * NLM + q] = s * aip + c * aiq;
      }
      __syncthreads();
      // A <- J^T * A (disjoint row pairs)
      for (int it = 0; it < 128; ++it) {
        int idx = it * BLK + tid;
        int k = idx >> 8, j = idx & 255;
        int p = sp[k], q = sqi[k];
        float c = sc[k], s = ss[k];
        float apj = A[p * NLM + j], aqj = A[q * NLM + j];
        A[p * NLM + j] = c * apj - s * aqj;
        A[q * NLM + j] = s * apj + c * aqj;
      }
      __syncthreads();
    }
  }

  ev[tid] = A[tid * NLM + tid];
  __syncthreads();

  if (tid == 0) {
    float sig[NSIG];
    for (int k = 0; k < NSIG; ++k) {         // top-10 descending
      float bv = NEG_INF; int bi = 0;
      for (int i = 0; i < NLM; ++i)
        if (ev[i] > bv) { bv = ev[i]; bi = i; }
      sig[k] = bv;
      ev[bi] = NEG_INF;
    }
    float nrm = fabsf(sig[0]) + 1e-8f;
    float dsum = 0.0f, psum = 0.0f;
    for (int k = 0; k < NSIG; ++k) {
      sig[k] /= nrm;
      dsum += fabsf(sig[k] - prev_sig[k]);
      psum += fabsf(prev_sig[k]);
    }
    int first = (step_count[0] == 0);
    float rel = first ? 0.0f : dsum / (psum + 1e-8f);
    out[0] = (!first && rel > 0.2f) ? 1.0f : 0.0f;  // is_rupture
    out[1] = rel;
    for (int k = 0; k < NSIG; ++k) out[2 + k] = sig[k];
  }
}

// ---------------------------------------------------------------------------
extern "C" void kernel_launch(void* const* d_in, const int* in_sizes, int n_in,
                              void* d_out, int out_size, void* d_ws, size_t ws_size,
                              hipStream_t stream) {
  const float* points    = (const float*)d_in[0];
  const float* prev_sig  = (const float*)d_in[1];
  const int*   step_count = (const int*)d_in[2];
  float* out = (float*)d_out;

  int n  = in_sizes[0] / DIM;          // 500000
  int nb = (n + BLK - 1) / BLK;        // FPS sweep blocks

  // workspace layout (256B-aligned sections)
  char* ws = (char*)d_ws;
  size_t off = 0;
  auto take = [&](size_t bytes) { char* p = ws + off; off += (bytes + 255) & ~(size_t)255; return p; };
  float* dists = (float*)take((size_t)n * sizeof(float));
  float* pmax  = (float*)take((size_t)nb * sizeof(float));
  int*   pidx  = (int*)  take((size_t)nb * sizeof(int));
  float* lm    = (float*)take((size_t)NLM * DIM * sizeof(float));
  float* Dmat  = (float*)take((size_t)NLM * NLM * sizeof(float));
  (void)ws_size; (void)n_in; (void)out_size;

  // Stage 1: FPS (255 sequential L2-resident sweeps)
  fps_init_kernel<<<nb, BLK, 0, stream>>>(points, n, dists, lm);
  for (int k = 0; k < NLM - 1; ++k) {
    fps_update_kernel<<<nb, BLK, 0, stream>>>(points, n, lm + k * DIM,
                                              dists, pmax, pidx);
    fps_select_kernel<<<1, BLK, 0, stream>>>(points, pmax, pidx, nb,
                                             lm + (k + 1) * DIM);
  }

  // Stage 2: distance matrix via WMMA
  gram_wmma_kernel<<<1, BLK, (NLM * DIM + NLM) * sizeof(float), stream>>>(lm, Dmat);

  // Stage 3: Jacobi eigensolve in LDS + signature/rupture outputs
  jacobi_kernel<<<1, BLK, NLM * NLM * sizeof(float), stream>>>(Dmat, prev_sig,
                                                               step_count, out);
}